// Edge_12017318494686
// MI455X (gfx1250) — compile-verified
//
#include <hip/hip_runtime.h>

// ---------------------------------------------------------------------------
// 8-direction edge detector, MI455X (gfx1250, wave32).
// Store-bandwidth-bound: 403 MB out + 50 MB in @ 23.3 TB/s => ~19.5 us floor.
// Strategy: LDS-tile the input with a +-2 halo (1.0-padded borders), one
// thread computes a 1x4 pixel strip so all 8 channel outputs are 16B
// non-temporal global_store_b128. Input staging uses gfx1250 async
// global->LDS copies (ASYNCcnt-tracked) — builtin params are int*-typed.
// ---------------------------------------------------------------------------

#define AS1 __attribute__((address_space(1)))
#define AS3 __attribute__((address_space(3)))

#if defined(__gfx1250__) && __has_builtin(__builtin_amdgcn_global_load_async_to_lds_b32) && __has_builtin(__builtin_amdgcn_s_wait_asynccnt)
#define USE_ASYNC_LDS 1
#else
#define USE_ASYNC_LDS 0
#endif

typedef float v4f __attribute__((ext_vector_type(4)));

namespace {
constexpr int Himg = 512;
constexpr int Wimg = 512;
constexpr int TX   = 128;   // tile pixels in x (32 threads * 4 px)
constexpr int TY   = 8;     // tile pixels in y
constexpr int HALO = 2;     // stencil reach (dy in [-2,2], dx in [-2,1])
constexpr int LSTR  = TX + 2 * HALO;   // 132 floats per LDS row
constexpr int LROWS = TY + 2 * HALO;   // 12 rows
constexpr int TILE_ELEMS = LSTR * LROWS;  // 1584 floats = 6336 B
}  // namespace

__global__ __launch_bounds__(256) void edge8_kernel(const float* __restrict__ in,
                                                    float* __restrict__ out) {
  __shared__ float tile[TILE_ELEMS];

  const int tx = threadIdx.x;          // 0..31
  const int ty = threadIdx.y;          // 0..7
  const int tid = ty * 32 + tx;        // 0..255 (8 wave32s)
  const int bx = blockIdx.x;
  const int by = blockIdx.y;
  const int n  = blockIdx.z;           // plane: B*C = 48
  const int gx0 = bx * TX;
  const int gy0 = by * TY;
  const size_t plane = (size_t)n * (size_t)(Himg * Wimg);

  // ---- Stage input tile (+halo) into LDS; out-of-image halo = 1.0f ----
  for (int i = tid; i < TILE_ELEMS; i += 256) {
    const int ly = i / LSTR;
    const int lx = i - ly * LSTR;
    const int gy = gy0 + ly - HALO;
    const int gx = gx0 + lx - HALO;
    const bool inb = (gy >= 0) & (gy < Himg) & (gx >= 0) & (gx < Wimg);
#if USE_ASYNC_LDS
    if (inb) {
      // gfx1250 async copy: GLOBAL_LOAD_ASYNC_TO_LDS_B32 (ASYNCcnt-tracked)
      __builtin_amdgcn_global_load_async_to_lds_b32(
          (AS1 int*)(in + plane + (size_t)gy * Wimg + gx),
          (AS3 int*)&tile[i], /*offset=*/0, /*cpol=*/0);
    } else {
      tile[i] = 1.0f;  // DS store, DScnt-tracked
    }
#else
    tile[i] = inb ? in[plane + (size_t)gy * Wimg + gx] : 1.0f;
#endif
  }
#if USE_ASYNC_LDS
  __builtin_amdgcn_s_wait_asynccnt(0);
#endif
  __syncthreads();

  // ---- Per-thread 1x4 output strip ----
  const int oy    = gy0 + ty;
  const int xbase = gx0 + tx * 4;
  const int lyc   = ty + HALO;
  const int lxc0  = tx * 4 + HALO;

  const bool okT = (oy >= 2);
  const bool okB = (oy < Himg - 2);

  float o[8][4];
#pragma unroll
  for (int j = 0; j < 4; ++j) {
    const int lx = lxc0 + j;
    const int ox = xbase + j;
#define T(dy, dx) tile[(lyc + (dy)) * LSTR + (lx + (dx))]
    const float c     = T(0, 0);
    const float gm2m2 = T(-2, -2), gm20 = T(-2, 0);
    const float gm1m1 = T(-1, -1), gm10 = T(-1, 0), gm1p1 = T(-1, 1);
    const float g0m2  = T(0, -2),  g0m1 = T(0, -1), g0p1  = T(0, 1);
    const float gp1m1 = T(1, -1),  gp10 = T(1, 0),  gp1p1 = T(1, 1);
    const float gp2m2 = T(2, -2);
#undef T

    // central differences (padded-image coords)
    const float d11   = gm1m1 - c;   // roll(1,1)
    const float dnn11 = gp1p1 - c;   // roll(-1,-1)
    const float d01   = g0m1  - c;   // roll(0,1)
    const float dn01  = g0p1  - c;   // roll(0,-1)
    const float d10   = gm10  - c;   // roll(1,0)
    const float dn10  = gp10  - c;   // roll(-1,0)
    const float dn11  = gp1m1 - c;   // roll(-1,1)
    const float d1n1  = gm1p1 - c;   // roll(1,-1)

    const float a11  = fabsf(d11),  ann11 = fabsf(dnn11);
    const float a01  = fabsf(d01),  an01  = fabsf(dn01);
    const float a10  = fabsf(d10),  an10  = fabsf(dn10);
    const float an11 = fabsf(dn11), a1n1  = fabsf(d1n1);

    // c10 (NOTE: reference duplicates the r(diff_01,(1,0)) term -> weight 2)
    const int c10 = (int)(a10 > a01) + (int)(a10 > an01) + (int)(a10 > an10)
                  + 2 * (int)(a10 > fabsf(gm1m1 - gm10))
                  +     (int)(a10 > fabsf(gm1p1 - gm10));
    const bool e10  = (c10 > 4) && (d10 > 0.0f);
    const bool e10n = (c10 > 4) && (d10 < 0.0f);

    const int c01 = (int)(a01 > a10) + (int)(a01 > an10) + (int)(a01 > an01)
                  + (int)(a01 > fabsf(gm1m1 - g0m1))
                  + (int)(a01 > fabsf(gp1m1 - g0m1))
                  + (int)(a01 > fabsf(g0m2 - g0m1));
    const bool e01  = (c01 > 4) && (d01 > 0.0f);
    const bool e01n = (c01 > 4) && (d01 < 0.0f);

    const int c11 = (int)(a11 > an11) + (int)(a11 > a1n1) + (int)(a11 > ann11)
                  + (int)(a11 > fabsf(g0m2  - gm1m1))
                  + (int)(a11 > fabsf(gm20  - gm1m1))
                  + (int)(a11 > fabsf(gm2m2 - gm1m1));
    const bool e11  = (c11 > 4) && (d11 > 0.0f);
    const bool e11n = (c11 > 4) && (d11 < 0.0f);

    // cn11 (NOTE: reference duplicates the r(diff_n11,(-1,1)) term -> weight 2)
    const int cn11 = (int)(an11 > a11) + (int)(an11 > a1n1) + (int)(an11 > ann11)
                   +     (int)(an11 > fabsf(g0m2  - gp1m1))
                   + 2 * (int)(an11 > fabsf(gp2m2 - gp1m1));
    const bool en11  = (cn11 > 4) && (dn11 > 0.0f);
    const bool en11n = (cn11 > 4) && (dn11 < 0.0f);

    const bool okL = (ox >= 2);
    const bool okR = (ox < Wimg - 2);

    o[0][j] = (e10   && okT)        ? 1.0f : 0.0f;
    o[1][j] = (e10n  && okB)        ? 1.0f : 0.0f;
    o[2][j] = (e01   && okL)        ? 1.0f : 0.0f;
    o[3][j] = (e01n  && okR)        ? 1.0f : 0.0f;
    o[4][j] = (e11   && okT && okL) ? 1.0f : 0.0f;
    o[5][j] = (e11n  && okB && okR) ? 1.0f : 0.0f;
    o[6][j] = (en11  && okB && okL) ? 1.0f : 0.0f;
    o[7][j] = (en11n && okT && okR) ? 1.0f : 0.0f;
  }

  // ---- 8 non-temporal 16B stores: out[((n*8+k)*H + oy)*W + xbase] ----
  const size_t obase = ((size_t)n * 8 * Himg + (size_t)oy) * Wimg + (size_t)xbase;
#pragma unroll
  for (int k = 0; k < 8; ++k) {
    v4f v = {o[k][0], o[k][1], o[k][2], o[k][3]};
    __builtin_nontemporal_store(v, (v4f*)(out + obase + (size_t)k * (Himg * Wimg)));
  }
}

extern "C" void kernel_launch(void* const* d_in, const int* in_sizes, int n_in,
                              void* d_out, int out_size, void* d_ws, size_t ws_size,
                              hipStream_t stream) {
  const float* x = (const float*)d_in[0];
  float* out = (float*)d_out;
  const int planes = in_sizes[0] / (Himg * Wimg);  // B*C = 48
  dim3 block(32, 8, 1);                            // 8 wave32s
  dim3 grid(Wimg / TX, Himg / TY, planes);         // (4, 64, 48)
  edge8_kernel<<<grid, block, 0, stream>>>(x, out);
}